// GCN_32255204393116
// MI455X (gfx1250) — compile-verified
//
#include <hip/hip_runtime.h>
#include <math.h>

// ---------------------------------------------------------------------------
// GCNII (GCN2Conv x4) inference for MI455X / gfx1250, fp32 end-to-end.
//   h = relu(x @ w_lin + b);  x0 = h
//   repeat 4x: agg = A@h (edge gather + atomic scatter-add, L2 resident)
//              z = 0.9*agg + 0.1*x0  (x0 pre-folded into agg init)
//              h = relu((1-beta)*z + beta*(z @ conv_w[l]))
// Dense math uses V_WMMA_F32_16X16X4_F32 (wave32, 16x16 tile, K-step 4).
// ---------------------------------------------------------------------------

typedef __attribute__((ext_vector_type(2))) float v2f;
typedef __attribute__((ext_vector_type(8))) float v8f;

__device__ __forceinline__ v8f wmma_f32_16x16x4(v2f a, v2f b, v8f c) {
  // 8 args: (neg_a, A, neg_b, B, c_mod, C, reuse_a, reuse_b)
  return __builtin_amdgcn_wmma_f32_16x16x4_f32(false, a, false, b, (short)0, c,
                                               false, false);
}

// ---------------------------------------------------------------------------
// Kernel 1: h = relu(x @ w_lin + b_lin);  also writes x0 = h.
// Grid: N/16 blocks of 128 threads (4 waves). Wave w computes cols [16w,16w+16).
// x tile [16 x 128] staged in LDS with stride 132 (bank-conflict-free A reads).
// ---------------------------------------------------------------------------
__global__ void __launch_bounds__(128)
lin_gemm_relu(const float* __restrict__ x, const float* __restrict__ W,
              const float* __restrict__ bias, float* __restrict__ h,
              float* __restrict__ x0) {
  __shared__ float xt[16 * 132];

  const int row0 = blockIdx.x * 16;
  const int t = threadIdx.x;

  // Stage 16x128 fp32 tile: 8 threads per row, 16 floats (4x float4) each.
  {
    const int r = t >> 3;
    const int c0 = (t & 7) * 16;
    const float4* xp =
        (const float4*)(x + (size_t)(row0 + r) * 128 + c0);
#pragma unroll
    for (int j = 0; j < 4; ++j) {
      float4 v = xp[j];
      float* dst = &xt[r * 132 + c0 + 4 * j];
      dst[0] = v.x; dst[1] = v.y; dst[2] = v.z; dst[3] = v.w;
    }
  }
  __syncthreads();

  const int lane  = t & 31;
  const int wave  = t >> 5;
  const int lhalf = lane >> 4;   // 0: lanes 0-15, 1: lanes 16-31
  const int lm    = lane & 15;   // M index (A) / N index (B)
  const int nc0   = wave * 16;   // output column base for this wave

  v8f acc = {};
#pragma unroll
  for (int k0 = 0; k0 < 128; k0 += 4) {
    // A frag (16x4): lanes 0-15 carry K=k0,k0+1 ; lanes 16-31 carry K=k0+2,k0+3
    v2f a;
    const float* ap = &xt[lm * 132 + k0 + 2 * lhalf];
    a.x = ap[0];
    a.y = ap[1];
    // B frag (4x16) from row-major W[128][64]
    v2f b;
    const float* wp = W + (size_t)(k0 + 2 * lhalf) * 64 + nc0 + lm;
    b.x = wp[0];
    b.y = wp[64];
    acc = wmma_f32_16x16x4(a, b, acc);
  }

  const float bv = bias[nc0 + lm];
#pragma unroll
  for (int i = 0; i < 8; ++i) {
    const int m = i + 8 * lhalf;         // C/D layout: VGPR i -> rows i, i+8
    float v = acc[i] + bv;
    v = fmaxf(v, 0.0f);
    const size_t idx = (size_t)(row0 + m) * 64 + nc0 + lm;
    h[idx] = v;
    x0[idx] = v;
  }
}

// ---------------------------------------------------------------------------
// Kernel 2: agg = (ALPHA/(1-ALPHA)) * x0   (residual pre-fold; float4 fill)
// ---------------------------------------------------------------------------
__global__ void init_agg(const float* __restrict__ x0, float* __restrict__ agg,
                         int n4) {
  const int i = blockIdx.x * blockDim.x + threadIdx.x;
  if (i >= n4) return;
  const float s = 0.1f / 0.9f;           // ALPHA/(1-ALPHA)
  float4 v = ((const float4*)x0)[i];
  v.x *= s; v.y *= s; v.z *= s; v.w *= s;
  ((float4*)agg)[i] = v;
}

// ---------------------------------------------------------------------------
// Kernel 3: SpMM scatter. One wave32 per edge; lane handles 2 features.
// Gather of h[col] is a coalesced 256B read (L2-resident); scatter uses
// global_atomic_add_f32 without return (STOREcnt path).
// ---------------------------------------------------------------------------
__global__ void __launch_bounds__(256)
spmm_scatter(const float* __restrict__ h, const int* __restrict__ row,
             const int* __restrict__ col, const float* __restrict__ ew,
             float* __restrict__ agg, int E) {
  const int wid = blockIdx.x * (blockDim.x >> 5) + (threadIdx.x >> 5);
  if (wid >= E) return;
  const int lane = threadIdx.x & 31;

  const int r = row[wid];
  const int c = col[wid];
  const float w = ew[wid];

  const float2 v = ((const float2*)(h + (size_t)c * 64))[lane];
  float* ap = agg + (size_t)r * 64 + 2 * lane;
  atomicAdd(ap + 0, v.x * w);
  atomicAdd(ap + 1, v.y * w);
}

// ---------------------------------------------------------------------------
// Kernel 4: z = 0.9*agg (x0 already folded in); h = relu((1-beta)z + beta*(z@W))
// Grid: N/16 blocks of 128 threads. z tile [16 x 64] in LDS, stride 68.
// ---------------------------------------------------------------------------
__global__ void __launch_bounds__(128)
gcn2_update(const float* __restrict__ agg, const float* __restrict__ W,
            float* __restrict__ out, float beta) {
  __shared__ float zt[16 * 68];

  const int row0 = blockIdx.x * 16;
  const int t = threadIdx.x;

  // Stage z tile: 8 threads per row, 8 floats (2x float4) each, scaled by 0.9.
  {
    const int r = t >> 3;
    const int c0 = (t & 7) * 8;
    const float4* ap =
        (const float4*)(agg + (size_t)(row0 + r) * 64 + c0);
#pragma unroll
    for (int j = 0; j < 2; ++j) {
      float4 v = ap[j];
      float* dst = &zt[r * 68 + c0 + 4 * j];
      dst[0] = 0.9f * v.x; dst[1] = 0.9f * v.y;
      dst[2] = 0.9f * v.z; dst[3] = 0.9f * v.w;
    }
  }
  __syncthreads();

  const int lane  = t & 31;
  const int wave  = t >> 5;
  const int lhalf = lane >> 4;
  const int lm    = lane & 15;
  const int nc0   = wave * 16;

  v8f acc = {};
#pragma unroll
  for (int k0 = 0; k0 < 64; k0 += 4) {
    v2f a;
    const float* ap = &zt[lm * 68 + k0 + 2 * lhalf];
    a.x = ap[0];
    a.y = ap[1];
    v2f b;
    const float* wp = W + (size_t)(k0 + 2 * lhalf) * 64 + nc0 + lm;
    b.x = wp[0];
    b.y = wp[64];
    acc = wmma_f32_16x16x4(a, b, acc);
  }

  const float omb = 1.0f - beta;
#pragma unroll
  for (int i = 0; i < 8; ++i) {
    const int m = i + 8 * lhalf;
    const float z = zt[m * 68 + nc0 + lm];
    const float v = omb * z + beta * acc[i];
    out[(size_t)(row0 + m) * 64 + nc0 + lm] = fmaxf(v, 0.0f);
  }
}

// ---------------------------------------------------------------------------
// Host-side orchestration (graph-capture safe: launches only, all on stream).
// Inputs: x[N*128], row[E], col[E], edge_weight[E], w_lin[128*64], b_lin[64],
//         conv_w[4*64*64].  Output: h_final [N*64] fp32.
// Workspace: h | x0 | agg  (3 * N*64 floats = ~76.8 MB)
// ---------------------------------------------------------------------------
extern "C" void kernel_launch(void* const* d_in, const int* in_sizes, int n_in,
                              void* d_out, int out_size, void* d_ws,
                              size_t ws_size, hipStream_t stream) {
  const float* x     = (const float*)d_in[0];
  const int*   row   = (const int*)d_in[1];
  const int*   col   = (const int*)d_in[2];
  const float* ew    = (const float*)d_in[3];
  const float* w_lin = (const float*)d_in[4];
  const float* b_lin = (const float*)d_in[5];
  const float* convw = (const float*)d_in[6];

  const int N = in_sizes[0] / 128;   // 100000 (multiple of 16)
  const int E = in_sizes[1];         // 1600000 (multiple of 8)
  const int D = 64;

  float* ws  = (float*)d_ws;
  float* h   = ws;
  float* x0  = h + (size_t)N * D;
  float* agg = x0 + (size_t)N * D;
  float* outf = (float*)d_out;

  const int nblk = N / 16;           // 6250 row-tiles
  const int n4 = N * D / 4;

  lin_gemm_relu<<<nblk, 128, 0, stream>>>(x, w_lin, b_lin, h, x0);

  for (int l = 0; l < 4; ++l) {
    const float beta = logf(0.5f / (float)(l + 1) + 1.0f);
    init_agg<<<(n4 + 255) / 256, 256, 0, stream>>>(x0, agg, n4);
    spmm_scatter<<<(E + 7) / 8, 256, 0, stream>>>(h, row, col, ew, agg, E);
    gcn2_update<<<nblk, 128, 0, stream>>>(
        agg, convw + (size_t)l * D * D, (l == 3) ? outf : h, beta);
  }
}